// Custom_Loss_1_42545946034230
// MI455X (gfx1250) — compile-verified
//
#include <hip/hip_runtime.h>

typedef float v2f __attribute__((ext_vector_type(2)));
typedef float v8f __attribute__((ext_vector_type(8)));

#define BLOCKS  2048
#define THREADS 256
#define WAVES_PER_BLOCK (THREADS / 32)

// Pass 1: each wave processes 16-sample x 16-feature tiles.
// A = w[m] * |out - target| (f32, 16x4 per wmma step), B = ones.
// C accumulates sum_f w[m]*|diff[m][f]| per sample row, replicated over N.
__global__ void l1_weighted_wmma_kernel(const float* __restrict__ out,
                                        const float* __restrict__ tgt,
                                        const float* __restrict__ x,
                                        float* __restrict__ partials,
                                        int numTiles)
{
    __shared__ float waveSums[WAVES_PER_BLOCK];

    const int tid  = threadIdx.x;
    const int lane = tid & 31;
    const int wid  = tid >> 5;
    const int globalWave = blockIdx.x * WAVES_PER_BLOCK + wid;
    const int totalWaves = gridDim.x * WAVES_PER_BLOCK;

    const int m = lane & 15;   // sample row within tile (A-matrix M)
    const int g = lane >> 4;   // half-wave selects K pair {0,1} vs {2,3}

    v8f c = {0.f, 0.f, 0.f, 0.f, 0.f, 0.f, 0.f, 0.f};
    v2f bones;
    bones[0] = 1.0f;
    bones[1] = 1.0f;

    for (int t = globalWave; t < numTiles; t += totalWaves) {   // wave-uniform loop: EXEC stays full
        const int s = t * 16 + m;
        const float w = 1.0f + 0.1f * x[s * 8 + 3];
        const float* __restrict__ orow = out + (size_t)s * 16;
        const float* __restrict__ trow = tgt + (size_t)s * 16;
#pragma unroll
        for (int kk = 0; kk < 4; ++kk) {
            const int off = kk * 4 + g * 2;           // lane's float2 within the 64B row
            v2f o  = *(const v2f*)(orow + off);       // global_load_b64, coalesced across wave
            v2f tv = *(const v2f*)(trow + off);
            v2f a;
            a[0] = w * __builtin_fabsf(o[0] - tv[0]); // A[m][K]   (K = g*2 + 0 of this step)
            a[1] = w * __builtin_fabsf(o[1] - tv[1]); // A[m][K+1]
            // D = A x ones + C  ->  row sums accumulate into C, replicated over all N
            c = __builtin_amdgcn_wmma_f32_16x16x4_f32(
                    false, a, false, bones, (short)0, c, false, false);
        }
    }

    // Each lane: sum its 8 C registers. Lanes 0-15 all hold total(samples 0-7),
    // lanes 16-31 all hold total(samples 8-15) -> wave reduce over-counts x16.
    float laneSum = c[0] + c[1] + c[2] + c[3] + c[4] + c[5] + c[6] + c[7];
#pragma unroll
    for (int offs = 16; offs >= 1; offs >>= 1)
        laneSum += __shfl_xor(laneSum, offs, 32);

    if (lane == 0) waveSums[wid] = laneSum * (1.0f / 16.0f);   // undo N-replication
    __syncthreads();

    if (tid == 0) {
        float s = 0.f;
        for (int i = 0; i < WAVES_PER_BLOCK; ++i) s += waveSums[i];  // fixed order: deterministic
        partials[blockIdx.x] = s;
    }
}

// Pass 2: single block, deterministic tree reduce of block partials (+ generic tail path),
// then apply mean scaling 1/(16*B).
__global__ void l1_finalize_kernel(const float* __restrict__ partials, int numParts,
                                   const float* __restrict__ out,
                                   const float* __restrict__ tgt,
                                   const float* __restrict__ x,
                                   int tailStart, int Btotal,
                                   float* __restrict__ result)
{
    __shared__ float sh[THREADS];
    const int tid = threadIdx.x;

    float s = 0.f;
    for (int i = tid; i < numParts; i += THREADS) s += partials[i];   // fixed stride order

    // Tail samples not covered by full 16-row tiles (none when B % 16 == 0).
    for (int si = tailStart + tid; si < Btotal; si += THREADS) {
        float acc = 0.f;
#pragma unroll
        for (int f = 0; f < 16; ++f)
            acc += __builtin_fabsf(out[(size_t)si * 16 + f] - tgt[(size_t)si * 16 + f]);
        s += acc * (1.0f + 0.1f * x[si * 8 + 3]);
    }

    sh[tid] = s;
    __syncthreads();
#pragma unroll
    for (int stride = THREADS / 2; stride >= 1; stride >>= 1) {
        if (tid < stride) sh[tid] += sh[tid + stride];
        __syncthreads();
    }
    if (tid == 0)
        result[0] = sh[0] * (1.0f / (16.0f * (float)Btotal));
}

extern "C" void kernel_launch(void* const* d_in, const int* in_sizes, int n_in,
                              void* d_out, int out_size, void* d_ws, size_t ws_size,
                              hipStream_t stream)
{
    (void)n_in; (void)out_size; (void)ws_size;

    const float* out_p = (const float*)d_in[0];   // [B,16] f32
    const float* tgt_p = (const float*)d_in[1];   // [B,16] f32
    const float* x_p   = (const float*)d_in[2];   // [B,8]  f32

    const int B         = in_sizes[0] / 16;
    const int numTiles  = B / 16;
    const int tailStart = numTiles * 16;

    float* partials = (float*)d_ws;               // BLOCKS floats of scratch

    l1_weighted_wmma_kernel<<<BLOCKS, THREADS, 0, stream>>>(
        out_p, tgt_p, x_p, partials, numTiles);

    l1_finalize_kernel<<<1, THREADS, 0, stream>>>(
        partials, BLOCKS, out_p, tgt_p, x_p, tailStart, B, (float*)d_out);
}